// Encoder_59854664237737
// MI455X (gfx1250) — compile-verified
//
#include <hip/hip_runtime.h>

// ---------------------------------------------------------------------------
// Problem constants (from reference)
// ---------------------------------------------------------------------------
#define NU_   200000
#define NI_   100000
#define DU_   64
#define DI_   128
#define H_    128
#define OUT_  64
#define E_UB_ 1000000
#define E_IU_ 1000000
#define E_II_ 500000

#define FLAG_ACC  1
#define FLAG_BIAS 2

typedef __attribute__((ext_vector_type(2))) float v2f;
typedef __attribute__((ext_vector_type(8))) float v8f;

// ---------------------------------------------------------------------------
// Degree count + reciprocal
// ---------------------------------------------------------------------------
__global__ void count_kernel(const int* __restrict__ dst, float* __restrict__ cnt, int E) {
    int i = blockIdx.x * blockDim.x + threadIdx.x;
    int stride = gridDim.x * blockDim.x;
    for (; i < E; i += stride)
        atomicAdd(&cnt[dst[i]], 1.0f);
}

__global__ void inv_kernel(float* __restrict__ cnt, int n) {
    int i = blockIdx.x * blockDim.x + threadIdx.x;
    int stride = gridDim.x * blockDim.x;
    for (; i < n; i += stride)
        cnt[i] = 1.0f / fmaxf(cnt[i], 1.0f);
}

// ---------------------------------------------------------------------------
// Edge scatter: one wave (32 lanes) per edge, atomic-add full feature row
// ---------------------------------------------------------------------------
__global__ void scatter_add_kernel(const float* __restrict__ xsrc,
                                   const int* __restrict__ src,
                                   const int* __restrict__ dst,
                                   float* __restrict__ agg, int E, int D) {
    int wid  = (blockIdx.x * blockDim.x + threadIdx.x) >> 5;
    int lane = threadIdx.x & 31;
    int nw   = (gridDim.x * blockDim.x) >> 5;
    for (int e = wid; e < E; e += nw) {
        int s = src[e];
        int d = dst[e];
        const float* sp = xsrc + (size_t)s * D;
        float*       dp = agg  + (size_t)d * D;
        for (int v = lane; v < D; v += 32)
            atomicAdd(dp + v, sp[v]);
    }
}

// ---------------------------------------------------------------------------
// FP32 WMMA GEMM:  C[M,N] (+)= (rowscale? diag(rowscale):I) * A[M,K] @ W[K,N] (+ bias)
// Block: 256 threads = 8 wave32s; block tile 128 rows x NCOLS cols.
// A tile in LDS stride 36 (16B-aligned rows, conflict-free column reads).
// W tile in LDS K-pair interleaved: (k,n) -> Ws[(k/2)*PSTRIDE + 2n + (k&1)],
// so one B fragment = one aligned ds_load_b64 (no VGPR repacking movs).
// V_WMMA_F32_16X16X4_F32, K chunked by 32.
// ---------------------------------------------------------------------------
template <int NCOLS>
__global__ __launch_bounds__(256)
void gemm_wmma_kernel(const float* __restrict__ A, const float* __restrict__ W,
                      const float* __restrict__ bias, const float* __restrict__ rowscale,
                      float* __restrict__ C, int M, int K, int flags) {
    constexpr int NTILES  = NCOLS / 16;
    constexpr int ASTRIDE = 36;              // floats per A tile row (16B aligned)
    constexpr int PSTRIDE = 2 * NCOLS + 8;   // floats per W pair-row (16B aligned)

    __shared__ float As[128 * ASTRIDE];
    __shared__ float Ws[16 * PSTRIDE];

    const int tid     = threadIdx.x;
    const int lane    = tid & 31;
    const int wave    = tid >> 5;
    const int rowbase = blockIdx.x * 128;
    const int m0      = wave * 16;       // wave's 16-row slice within block tile
    const int half    = lane >> 4;       // 0: lanes 0-15, 1: lanes 16-31
    const int l16     = lane & 15;

    v8f acc[NTILES];
    v8f zero = {};
#pragma unroll
    for (int t = 0; t < NTILES; ++t) acc[t] = zero;

    for (int kb = 0; kb < K; kb += 32) {
        // ---- Stage A chunk [128 x 32] as float4s (optional 1/deg row scale) ----
#pragma unroll
        for (int j = 0; j < 4; ++j) {
            int idx = tid + j * 256;         // 1024 float4 slots
            int m   = idx >> 3;
            int q   = idx & 7;
            int gr  = rowbase + m;
            float4 v = make_float4(0.0f, 0.0f, 0.0f, 0.0f);
            if (gr < M) {
                v = *(const float4*)(A + (size_t)gr * K + kb + 4 * q);
                if (rowscale) {
                    float sc = rowscale[gr];
                    v.x *= sc; v.y *= sc; v.z *= sc; v.w *= sc;
                }
            }
            *(float4*)(As + m * ASTRIDE + 4 * q) = v;
        }
        // ---- Stage W chunk [32 x NCOLS] K-pair interleaved ----
#pragma unroll
        for (int j = 0; j < (16 * (NCOLS / 2)) / 256; ++j) {
            int idx = tid + j * 256;         // quads: pair-row p, column pair n,n+1
            int p   = idx / (NCOLS / 2);
            int c   = idx - p * (NCOLS / 2);
            int n   = 2 * c;
            float2 g0 = *(const float2*)(W + (size_t)(kb + 2 * p) * NCOLS + n);
            float2 g1 = *(const float2*)(W + (size_t)(kb + 2 * p + 1) * NCOLS + n);
            float4 st;
            st.x = g0.x; st.y = g1.x; st.z = g0.y; st.w = g1.y;
            *(float4*)(Ws + p * PSTRIDE + 2 * n) = st;
        }
        __syncthreads();

#pragma unroll
        for (int k4 = 0; k4 < 32; k4 += 4) {
            // A fragment (16x4 f32): VGPR0 = K{k4, k4+2}, VGPR1 = K{k4+1, k4+3}
            int ka = k4 + half * 2;          // even
            v2f af = *(const v2f*)(As + (m0 + l16) * ASTRIDE + ka);
#pragma unroll
            for (int t = 0; t < NTILES; ++t) {
                v2f bf = *(const v2f*)(Ws + (ka >> 1) * PSTRIDE + 2 * (t * 16 + l16));
                acc[t] = __builtin_amdgcn_wmma_f32_16x16x4_f32(
                    false, af, false, bf, (short)0, acc[t], false, false);
            }
        }
        __syncthreads();
    }

    // Epilogue: C/D layout — VGPR j holds M = j (lanes 0-15) / j+8 (lanes 16-31)
#pragma unroll
    for (int t = 0; t < NTILES; ++t) {
#pragma unroll
        for (int j = 0; j < 8; ++j) {
            int r = rowbase + m0 + half * 8 + j;
            if (r < M) {
                int    col = t * 16 + l16;
                size_t off = (size_t)r * NCOLS + col;
                float  v   = acc[t][j];
                if (flags & FLAG_BIAS) v += bias[col];
                if (flags & FLAG_ACC)  v += C[off];
                C[off] = v;
            }
        }
    }
}

// ---------------------------------------------------------------------------
// Elementwise (float4): user relu, item combine xi = relu(0.5*(o_ub + o_ii))
// ---------------------------------------------------------------------------
__global__ void relu4_kernel(float4* __restrict__ x, size_t n4) {
    size_t i = (size_t)blockIdx.x * blockDim.x + threadIdx.x;
    size_t stride = (size_t)gridDim.x * blockDim.x;
    for (; i < n4; i += stride) {
        float4 v = x[i];
        v.x = fmaxf(v.x, 0.0f); v.y = fmaxf(v.y, 0.0f);
        v.z = fmaxf(v.z, 0.0f); v.w = fmaxf(v.w, 0.0f);
        x[i] = v;
    }
}

__global__ void combine4_kernel(const float4* __restrict__ a, float4* __restrict__ b, size_t n4) {
    size_t i = (size_t)blockIdx.x * blockDim.x + threadIdx.x;
    size_t stride = (size_t)gridDim.x * blockDim.x;
    for (; i < n4; i += stride) {
        float4 u = a[i];
        float4 v = b[i];
        v.x = fmaxf(0.5f * (u.x + v.x), 0.0f);
        v.y = fmaxf(0.5f * (u.y + v.y), 0.0f);
        v.z = fmaxf(0.5f * (u.z + v.z), 0.0f);
        v.w = fmaxf(0.5f * (u.w + v.w), 0.0f);
        b[i] = v;
    }
}

// ---------------------------------------------------------------------------
// Host-side launch helpers
// ---------------------------------------------------------------------------
static inline void run_gemm(const float* A, const float* W, const float* bias,
                            const float* rowscale, float* C,
                            int M, int K, int N, int flags, hipStream_t s) {
    dim3 grid((unsigned)((M + 127) / 128));
    if (N == 128) {
        gemm_wmma_kernel<128><<<grid, 256, 0, s>>>(A, W, bias, rowscale, C, M, K, flags);
    } else {
        gemm_wmma_kernel<64><<<grid, 256, 0, s>>>(A, W, bias, rowscale, C, M, K, flags);
    }
}

static inline void run_scatter(const float* xsrc, const int* src, const int* dst,
                               float* agg, int E, int D, hipStream_t s) {
    scatter_add_kernel<<<4096, 256, 0, s>>>(xsrc, src, dst, agg, E, D);
}

static inline void run_counts(const int* dst, float* cnt, int E, int n, hipStream_t s) {
    hipMemsetAsync(cnt, 0, (size_t)n * sizeof(float), s);
    count_kernel<<<2048, 256, 0, s>>>(dst, cnt, E);
    inv_kernel<<<(n + 255) / 256, 256, 0, s>>>(cnt, n);
}

// ---------------------------------------------------------------------------
// kernel_launch
// ---------------------------------------------------------------------------
extern "C" void kernel_launch(void* const* d_in, const int* in_sizes, int n_in,
                              void* d_out, int out_size, void* d_ws, size_t ws_size,
                              hipStream_t stream) {
    (void)in_sizes; (void)n_in; (void)out_size; (void)ws_size;

    const float* x_user = (const float*)d_in[0];
    const float* x_item = (const float*)d_in[1];
    const int* ub_src = (const int*)d_in[2];
    const int* ub_dst = (const int*)d_in[3];
    const int* iu_src = (const int*)d_in[4];
    const int* iu_dst = (const int*)d_in[5];
    const int* ii_src = (const int*)d_in[6];
    const int* ii_dst = (const int*)d_in[7];
    const float* W1l_ub = (const float*)d_in[8];
    const float* W1r_ub = (const float*)d_in[9];
    const float* b1_ub  = (const float*)d_in[10];
    const float* W1l_iu = (const float*)d_in[11];
    const float* W1r_iu = (const float*)d_in[12];
    const float* b1_iu  = (const float*)d_in[13];
    const float* W1l_ii = (const float*)d_in[14];
    const float* W1r_ii = (const float*)d_in[15];
    const float* b1_ii  = (const float*)d_in[16];
    const float* W23l   = (const float*)d_in[17];
    const float* W23r   = (const float*)d_in[18];
    const float* b23    = (const float*)d_in[19];
    const float* Wf_user = (const float*)d_in[20];
    const float* bf_user = (const float*)d_in[21];
    const float* Wf_item = (const float*)d_in[22];
    const float* bf_item = (const float*)d_in[23];
    float* outp = (float*)d_out;

    // Workspace layout (floats)
    float* ws = (float*)d_ws;
    const size_t oNUH = (size_t)NU_ * H_;
    const size_t oNIH = (size_t)NI_ * H_;
    float* xu_a  = ws;
    float* xu_b  = xu_a + oNUH;
    float* xi_a  = xu_b + oNUH;
    float* xi_b  = xi_a + oNIH;
    float* Tbuf  = xi_b + oNIH;          // o_i_ub staging
    float* agg   = Tbuf + oNIH;          // max(NU,NI) * H
    float* inv_ub = agg + oNUH;          // NI
    float* inv_iu = inv_ub + NI_;        // NU
    float* inv_ii = inv_iu + NU_;        // NI

    // --- degree reciprocals (fixed topology; recomputed each call) ---
    run_counts(ub_dst, inv_ub, E_UB_, NI_, stream);
    run_counts(iu_dst, inv_iu, E_IU_, NU_, stream);
    run_counts(ii_dst, inv_ii, E_II_, NI_, stream);

    // =================== Layer 1 ===================
    // o_i_ub -> Tbuf : mean_{ub} x_user @ W1l_ub + b + x_item @ W1r_ub
    hipMemsetAsync(agg, 0, (size_t)NI_ * DU_ * sizeof(float), stream);
    run_scatter(x_user, ub_src, ub_dst, agg, E_UB_, DU_, stream);
    run_gemm(agg,    W1l_ub, b1_ub, inv_ub, Tbuf, NI_, DU_, H_, FLAG_BIAS, stream);
    run_gemm(x_item, W1r_ub, nullptr, nullptr, Tbuf, NI_, DI_, H_, FLAG_ACC, stream);

    // o_u -> xu_b : mean_{iu} x_item @ W1l_iu + b + x_user @ W1r_iu ; relu
    hipMemsetAsync(agg, 0, (size_t)NU_ * DI_ * sizeof(float), stream);
    run_scatter(x_item, iu_src, iu_dst, agg, E_IU_, DI_, stream);
    run_gemm(agg,    W1l_iu, b1_iu, inv_iu, xu_b, NU_, DI_, H_, FLAG_BIAS, stream);
    run_gemm(x_user, W1r_iu, nullptr, nullptr, xu_b, NU_, DU_, H_, FLAG_ACC, stream);
    relu4_kernel<<<4096, 256, 0, stream>>>((float4*)xu_b, oNUH / 4);

    // o_i_ii -> xi_b ; xi = relu(0.5*(Tbuf + xi_b))
    hipMemsetAsync(agg, 0, (size_t)NI_ * DI_ * sizeof(float), stream);
    run_scatter(x_item, ii_src, ii_dst, agg, E_II_, DI_, stream);
    run_gemm(agg,    W1l_ii, b1_ii, inv_ii, xi_b, NI_, DI_, H_, FLAG_BIAS, stream);
    run_gemm(x_item, W1r_ii, nullptr, nullptr, xi_b, NI_, DI_, H_, FLAG_ACC, stream);
    combine4_kernel<<<4096, 256, 0, stream>>>((const float4*)Tbuf, (float4*)xi_b, oNIH / 4);

    float* xu = xu_b; float* xu_n = xu_a;
    float* xi = xi_b; float* xi_n = xi_a;

    // =================== Layers 2-3 ===================
    for (int l = 0; l < 2; ++l) {
        const float* Wl0 = W23l + ((size_t)(l * 3 + 0)) * H_ * H_;
        const float* Wr0 = W23r + ((size_t)(l * 3 + 0)) * H_ * H_;
        const float* b0  = b23  + ((size_t)(l * 3 + 0)) * H_;
        const float* Wl1 = W23l + ((size_t)(l * 3 + 1)) * H_ * H_;
        const float* Wr1 = W23r + ((size_t)(l * 3 + 1)) * H_ * H_;
        const float* b1  = b23  + ((size_t)(l * 3 + 1)) * H_;
        const float* Wl2 = W23l + ((size_t)(l * 3 + 2)) * H_ * H_;
        const float* Wr2 = W23r + ((size_t)(l * 3 + 2)) * H_ * H_;
        const float* b2  = b23  + ((size_t)(l * 3 + 2)) * H_;

        // o_ub -> Tbuf (reads xu, xi)
        hipMemsetAsync(agg, 0, (size_t)NI_ * H_ * sizeof(float), stream);
        run_scatter(xu, ub_src, ub_dst, agg, E_UB_, H_, stream);
        run_gemm(agg, Wl0, b0, inv_ub, Tbuf, NI_, H_, H_, FLAG_BIAS, stream);
        run_gemm(xi,  Wr0, nullptr, nullptr, Tbuf, NI_, H_, H_, FLAG_ACC, stream);

        // o_u -> xu_n (reads xi, xu) ; relu
        hipMemsetAsync(agg, 0, (size_t)NU_ * H_ * sizeof(float), stream);
        run_scatter(xi, iu_src, iu_dst, agg, E_IU_, H_, stream);
        run_gemm(agg, Wl1, b1, inv_iu, xu_n, NU_, H_, H_, FLAG_BIAS, stream);
        run_gemm(xu,  Wr1, nullptr, nullptr, xu_n, NU_, H_, H_, FLAG_ACC, stream);
        relu4_kernel<<<4096, 256, 0, stream>>>((float4*)xu_n, oNUH / 4);

        // o_ii -> xi_n (reads xi) ; xi_n = relu(0.5*(Tbuf + xi_n))
        hipMemsetAsync(agg, 0, (size_t)NI_ * H_ * sizeof(float), stream);
        run_scatter(xi, ii_src, ii_dst, agg, E_II_, H_, stream);
        run_gemm(agg, Wl2, b2, inv_ii, xi_n, NI_, H_, H_, FLAG_BIAS, stream);
        run_gemm(xi,  Wr2, nullptr, nullptr, xi_n, NI_, H_, H_, FLAG_ACC, stream);
        combine4_kernel<<<4096, 256, 0, stream>>>((const float4*)Tbuf, (float4*)xi_n, oNIH / 4);

        float* t;
        t = xu; xu = xu_n; xu_n = t;
        t = xi; xi = xi_n; xi_n = t;
    }

    // =================== Final Linear per node type ===================
    run_gemm(xu, Wf_user, bf_user, nullptr, outp,                      NU_, H_, OUT_, FLAG_BIAS, stream);
    run_gemm(xi, Wf_item, bf_item, nullptr, outp + (size_t)NU_ * OUT_, NI_, H_, OUT_, FLAG_BIAS, stream);
}